// FastGuidedFilter_attention_3676492005844
// MI455X (gfx1250) — compile-verified
//
#include <hip/hip_runtime.h>

#define H 512
#define WD 512
#define NPLANES 64
#define REG 48          // staged region (32 output + 2*8 halo)
#define STRIDE 49       // LDS row stride (pad to break bank alignment)
#define OUTT 32         // output tile per workgroup
#define HALO 8
#define NBUF (REG * STRIDE)   // 2352 floats per LDS buffer

typedef __attribute__((ext_vector_type(2))) float v2f;
typedef __attribute__((ext_vector_type(8))) float v8f;

#if __has_builtin(__builtin_amdgcn_global_load_async_to_lds_b32)
#define USE_ASYNC_LDS 1
typedef __attribute__((address_space(1))) int* glb_ip;   // global src (as printed by clang)
typedef __attribute__((address_space(3))) int* lds_ip;   // LDS dst
#else
#define USE_ASYNC_LDS 0
#endif

__device__ __forceinline__ int imin(int a, int b) { return a < b ? a : b; }
__device__ __forceinline__ int imax(int a, int b) { return a > b ? a : b; }

// D = A(16x4, f32) * B(4x16, f32) + C   -- native CDNA5 f32 WMMA
__device__ __forceinline__ v8f wmma4(v2f a, v2f b, v8f c) {
    return __builtin_amdgcn_wmma_f32_16x16x4_f32(false, a, false, b, (short)0, c,
                                                 false, false);
}

// 9-tap band weight: 1 if |out - k| <= 4
__device__ __forceinline__ float bandw(int o, int k) {
    int d = o - k;
    return (d >= -4 && d <= 4) ? 1.0f : 0.0f;
}

// Separable 9x9 box-sum over the 48x48 LDS region via WMMA f32 16x16x4.
// Row pass: RF[u][c] = sum_k src[u][k] * w(c,k)  (all 48 rows, cols [lo,hi))
// Col pass: dst[r][c] = sum_k RF[k][c] * w(r,k)  (rows & cols in [lo,hi))
// wavei is wave-uniform (readfirstlane'd) so all tile control flow is scalar.
__device__ void box_wmma(const float* __restrict__ src, float* __restrict__ rf,
                         float* __restrict__ dst, int lo, int hi, int wavei, int lane) {
    const int nV   = (hi - lo + 15) >> 4;      // 16-wide output tiles per dim
    const int M    = lane & 15;                // A-row / B,C,D-column index
    const int kh   = (lane < 16) ? 0 : 2;      // K lo/hi half select (A and B frags)
    const int rsel = (lane < 16) ? 0 : 8;      // D-row half select

    // ---- row pass ----
    for (int j = wavei; j < 3 * nV; j += 8) {  // u-tiles {0,16,32} x v-tiles
        int u0 = 16 * (j / nV);
        int v0 = lo + 16 * (j % nV);
        v8f acc = {0.f, 0.f, 0.f, 0.f, 0.f, 0.f, 0.f, 0.f};
        int klo = imax(0, v0 - 4) & ~3;
        int khi = imin(REG - 1, v0 + 19) & ~3;
        const float* srow = src + (u0 + M) * STRIDE;
        for (int k0 = klo; k0 <= khi; k0 += 4) {
            v2f a, b;
            a.x = srow[k0 + kh];
            a.y = srow[k0 + kh + 1];
            b.x = bandw(v0 + M, k0 + kh);
            b.y = bandw(v0 + M, k0 + kh + 1);
            acc = wmma4(a, b, acc);
        }
        int col = v0 + M;
        if (col < REG) {
            #pragma unroll
            for (int r = 0; r < 8; ++r) rf[(u0 + rsel + r) * STRIDE + col] = acc[r];
        }
    }
    __syncthreads();

    // ---- column pass ----
    for (int j = wavei; j < nV * nV; j += 8) {
        int u0 = lo + 16 * (j / nV);
        int v0 = lo + 16 * (j % nV);
        v8f acc = {0.f, 0.f, 0.f, 0.f, 0.f, 0.f, 0.f, 0.f};
        int klo = imax(0, u0 - 4) & ~3;
        int khi = imin(REG - 1, u0 + 19) & ~3;
        int vr  = imin(v0 + M, STRIDE - 1);    // clamp reads into pad column
        for (int k0 = klo; k0 <= khi; k0 += 4) {
            v2f a, b;
            a.x = bandw(u0 + M, k0 + kh);
            a.y = bandw(u0 + M, k0 + kh + 1);
            b.x = rf[(k0 + kh) * STRIDE + vr];
            b.y = rf[(k0 + kh + 1) * STRIDE + vr];
            acc = wmma4(a, b, acc);
        }
        int col = v0 + M;
        if (col < REG) {
            #pragma unroll
            for (int r = 0; r < 8; ++r) {
                int row = u0 + rsel + r;
                if (row < REG) dst[row * STRIDE + col] = acc[r];
            }
        }
    }
    __syncthreads();
}

// -------- deterministic global sum of |a| + 1e-12 (two-stage tree) --------
__global__ __launch_bounds__(256) void fgf_reduce1(const float* __restrict__ a, int n,
                                                   double* __restrict__ part) {
    __shared__ double s[256];
    int tid = threadIdx.x;
    double acc = 0.0;
    for (long i = (long)blockIdx.x * 256 + tid; i < n; i += 256L * 1024)
        acc += (double)fabsf(a[i]) + 1e-12;
    s[tid] = acc;
    __syncthreads();
    for (int o = 128; o > 0; o >>= 1) {
        if (tid < o) s[tid] += s[tid + o];
        __syncthreads();
    }
    if (tid == 0) part[blockIdx.x] = s[0];
}

__global__ __launch_bounds__(256) void fgf_reduce2(double* __restrict__ part) {
    __shared__ double s[256];
    int tid = threadIdx.x;
    double acc = 0.0;
    for (int i = tid; i < 1024; i += 256) acc += part[i];
    s[tid] = acc;
    __syncthreads();
    for (int o = 128; o > 0; o >>= 1) {
        if (tid < o) s[tid] += s[tid + o];
        __syncthreads();
    }
    if (tid == 0) part[1024] = s[0];
}

// -------------------------- fused guided filter --------------------------
__global__ __launch_bounds__(256) void fgf_main(const float* __restrict__ gx,
                                                const float* __restrict__ gy,
                                                const float* __restrict__ ga,
                                                const double* __restrict__ Ssum,
                                                float* __restrict__ gout) {
    __shared__ float smem[13 * NBUF];   // ~122 KB of the WGP's 320 KB LDS
    float* inX = smem + 0 * NBUF;
    float* inY = smem + 1 * NBUF;
    float* inA = smem + 2 * NBUF;
    float* P   = smem + 3 * NBUF;
    float* RF  = smem + 4 * NBUF;
    float* AA  = smem + 11 * NBUF;
    float* BB  = smem + 12 * NBUF;
    // box buffers: smem + (5+f)*NBUF, f = 0..5

    const int tid   = threadIdx.x;
    const int lane  = tid & 31;
    const int wavei = __builtin_amdgcn_readfirstlane(tid >> 5);  // scalar wave id
    const int g0h   = (int)blockIdx.y * OUTT - HALO;
    const int g0w   = (int)blockIdx.x * OUTT - HALO;
    const long pbase = (long)blockIdx.z * (H * WD);
    const float invS = (float)(1.0 / Ssum[0]);

    // zero all of LDS: provides the zero padding for masked async loads and
    // keeps pad columns / unwritten corners finite for WMMA fragments.
    for (int i = tid; i < 13 * NBUF; i += 256) smem[i] = 0.0f;
    __syncthreads();

    // stage 48x48 region, zero-padded outside the image (== truncated windows)
#if USE_ASYNC_LDS
    for (int idx = tid; idx < REG * REG; idx += 256) {
        int u = idx / REG, v = idx % REG;
        int h = g0h + u, w = g0w + v;
        if (((unsigned)h < H) && ((unsigned)w < WD)) {
            long gi = pbase + (long)h * WD + w;
            int o = u * STRIDE + v;
            // DMA memory -> LDS (ASYNCcnt); masked-off lanes keep the zero fill
            __builtin_amdgcn_global_load_async_to_lds_b32((glb_ip)(gx + gi),
                                                          (lds_ip)(inX + o), 0, 0);
            __builtin_amdgcn_global_load_async_to_lds_b32((glb_ip)(gy + gi),
                                                          (lds_ip)(inY + o), 0, 0);
            __builtin_amdgcn_global_load_async_to_lds_b32((glb_ip)(ga + gi),
                                                          (lds_ip)(inA + o), 0, 0);
        }
    }
#if __has_builtin(__builtin_amdgcn_s_wait_asynccnt)
    __builtin_amdgcn_s_wait_asynccnt(0);
#else
    asm volatile("s_wait_asynccnt 0" ::: "memory");
#endif
    __syncthreads();
    // a := |a| + EPSS inside the image, 0 outside (in-place LDS transform)
    for (int idx = tid; idx < REG * REG; idx += 256) {
        int u = idx / REG, v = idx % REG, o = u * STRIDE + v;
        int h = g0h + u, w = g0w + v;
        bool in = ((unsigned)h < H) && ((unsigned)w < WD);
        float av = inA[o];
        inA[o] = in ? (fabsf(av) + 1e-12f) : 0.0f;
    }
#else
    for (int idx = tid; idx < REG * REG; idx += 256) {
        int u = idx / REG, v = idx % REG;
        int h = g0h + u, w = g0w + v;
        bool in = ((unsigned)h < H) && ((unsigned)w < WD);
        long gi = pbase + (long)h * WD + w;
        int o = u * STRIDE + v;
        inX[o] = in ? gx[gi] : 0.0f;
        inY[o] = in ? gy[gi] : 0.0f;
        inA[o] = in ? (fabsf(ga[gi]) + 1e-12f) : 0.0f;
    }
#endif
    __syncthreads();

    // six box-filtered product fields, valid on region [4,44)^2
    for (int f = 0; f < 6; ++f) {
        for (int idx = tid; idx < REG * REG; idx += 256) {
            int u = idx / REG, v = idx % REG, o = u * STRIDE + v;
            float xv = inX[o], yv = inY[o], av = inA[o];
            float p;
            switch (f) {
                case 0:  p = av;                     break;  // a
                case 1:  p = av * av * xv * yv;      break;  // a^2 x y
                case 2:  p = av * av * xv * invS;    break;  // t a x
                case 3:  p = av * yv;                break;  // a y
                case 4:  p = av * av * xv * xv;      break;  // a^2 x^2
                default: p = av * xv;                break;  // a x
            }
            P[o] = p;
        }
        __syncthreads();
        box_wmma(P, RF, smem + (5 + f) * NBUF, 4, 44, wavei, lane);
    }

    // A, b on [4,44)^2 (zero outside the image -> truncated second-stage box)
    const float* B0 = smem + 5 * NBUF;
    const float* B1 = smem + 6 * NBUF;
    const float* B2 = smem + 7 * NBUF;
    const float* B3 = smem + 8 * NBUF;
    const float* B4 = smem + 9 * NBUF;
    const float* B5 = smem + 10 * NBUF;
    for (int idx = tid; idx < 40 * 40; idx += 256) {
        int u = 4 + idx / 40, v = 4 + idx % 40, o = u * STRIDE + v;
        int h = g0h + u, w = g0w + v;
        float Av = 0.0f, Bv = 0.0f;
        if (((unsigned)h < H) && ((unsigned)w < WD)) {
            float ba = B0[o], ba2xy = B1[o], btax = B2[o];
            float bay = B3[o], ba2x2 = B4[o], bax = B5[o];
            int nh = imin(h + 4, H - 1) - imax(h - 4, 0) + 1;
            int nw = imin(w + 4, WD - 1) - imax(w - 4, 0) + 1;
            float Nf = (float)(nh * nw);
            // A = (box_a2xy - box_tax*box_ay) / (|box_a2x2 - box_tax*box_ax| + N*EPS)
            float den = fabsf(ba2x2 - btax * bax) + Nf * 1e-8f;
            Av = (ba2xy - btax * bay) / den;
            Bv = (bay - Av * bax) / ba;
        }
        AA[o] = Av;
        BB[o] = Bv;
    }
    __syncthreads();

    // second-stage box of A and b, valid on [8,40)^2 (reuse box buffers 0,1)
    float* BXA = smem + 5 * NBUF;
    float* BXB = smem + 6 * NBUF;
    box_wmma(AA, RF, BXA, 8, 40, wavei, lane);
    box_wmma(BB, RF, BXB, 8, 40, wavei, lane);

    // out = (boxA/N)*x + boxB/N
    for (int idx = tid; idx < OUTT * OUTT; idx += 256) {
        int u = 8 + idx / OUTT, v = 8 + idx % OUTT, o = u * STRIDE + v;
        int h = g0h + u, w = g0w + v;
        if (((unsigned)h < H) && ((unsigned)w < WD)) {
            int nh = imin(h + 4, H - 1) - imax(h - 4, 0) + 1;
            int nw = imin(w + 4, WD - 1) - imax(w - 4, 0) + 1;
            float rN = 1.0f / (float)(nh * nw);
            gout[pbase + (long)h * WD + w] = (BXA[o] * rN) * inX[o] + BXB[o] * rN;
        }
    }
}

extern "C" void kernel_launch(void* const* d_in, const int* in_sizes, int n_in,
                              void* d_out, int out_size, void* d_ws, size_t ws_size,
                              hipStream_t stream) {
    (void)n_in; (void)out_size; (void)ws_size;
    const float* gx = (const float*)d_in[0];
    const float* gy = (const float*)d_in[1];
    const float* ga = (const float*)d_in[2];
    float* gout = (float*)d_out;
    double* W = (double*)d_ws;          // [0..1023] partials, [1024] total
    int n = in_sizes[2];

    fgf_reduce1<<<1024, 256, 0, stream>>>(ga, n, W);
    fgf_reduce2<<<1, 256, 0, stream>>>(W);
    dim3 grid(WD / OUTT, H / OUTT, NPLANES);
    fgf_main<<<grid, 256, 0, stream>>>(gx, gy, ga, W + 1024, gout);
}